// TripletAttention_23596550324828
// MI455X (gfx1250) — compile-verified
//
#include <hip/hip_runtime.h>
#include <math.h>

// TripletAttention for x[32,256,64,64] fp32.
// k1: channel max+mean pool (float4 streaming) -> pmax/pavg [32,64,64] (ws)
// k2: 3x (7x7 conv 2->1 + BN + sigmoid) as one im2col GEMM per 16-pixel tile
//     via V_WMMA_F32_16X16X4_F32. K padded to 128 = 2ch x 8r x 8s so the
//     k->(ch,r,s) decode is pure bit ops and fully constant-folds under
//     unrolling (zero weights at r==7/s==7 taps; LDS tile pre-zeroed).
//     Branch 2's H<->W transpose == spatially transposed kernel taps.
// k3: out = x * att (broadcast over C), float4 streaming.

typedef __attribute__((ext_vector_type(2))) float v2f;
typedef __attribute__((ext_vector_type(8))) float v8f;

#define BATCH 32
#define CCH   256
#define HH    64
#define WW    64
#define HW    (HH * WW)          // 4096
#define CHW   (CCH * HW)         // 1<<20
#define EPSV  1e-5f

// ---------------- Kernel 1: pool over C ----------------
// One block per (b,h): 256 threads = 16 c-groups x 16 float4 columns.
__global__ void pool_kernel(const float* __restrict__ x,
                            float* __restrict__ pmax,
                            float* __restrict__ pavg) {
  __shared__ float4 smax[256];
  __shared__ float4 ssum[256];
  const int bh = blockIdx.x;            // b*64 + h
  const int b = bh >> 6, h = bh & 63;
  const int t = threadIdx.x;
  const int w4 = t & 15, cg = t >> 4;   // 16 float4 cols, 16 c-groups
  const float4* base = (const float4*)(x + (size_t)b * CHW + (size_t)h * WW) + w4;
  float4 mx = make_float4(-INFINITY, -INFINITY, -INFINITY, -INFINITY);
  float4 sm = make_float4(0.f, 0.f, 0.f, 0.f);
  const int c0 = cg * 16;
#pragma unroll 4
  for (int i = 0; i < 16; ++i) {
    float4 v = base[(size_t)(c0 + i) * (HW / 4)];
    mx.x = fmaxf(mx.x, v.x); mx.y = fmaxf(mx.y, v.y);
    mx.z = fmaxf(mx.z, v.z); mx.w = fmaxf(mx.w, v.w);
    sm.x += v.x; sm.y += v.y; sm.z += v.z; sm.w += v.w;
  }
  smax[t] = mx;
  ssum[t] = sm;
  __syncthreads();
  // log2 reduction over the 16 c-groups (stride in t is 16)
  for (int s = 128; s >= 16; s >>= 1) {
    if (t < s) {
      float4 a = smax[t], c = smax[t + s];
      a.x = fmaxf(a.x, c.x); a.y = fmaxf(a.y, c.y);
      a.z = fmaxf(a.z, c.z); a.w = fmaxf(a.w, c.w);
      smax[t] = a;
      float4 u = ssum[t], d = ssum[t + s];
      u.x += d.x; u.y += d.y; u.z += d.z; u.w += d.w;
      ssum[t] = u;
    }
    __syncthreads();
  }
  if (t < 16) {
    float4 m2 = smax[t];
    float4 s2 = ssum[t];
    s2.x *= (1.f / 256.f); s2.y *= (1.f / 256.f);
    s2.z *= (1.f / 256.f); s2.w *= (1.f / 256.f);
    ((float4*)pmax)[bh * 16 + t] = m2;
    ((float4*)pavg)[bh * 16 + t] = s2;
  }
}

// ---------------- Kernel 2: conv+BN+sigmoid x3 -> att ----------------
// One block per (b,h): 4 waves, wave wid handles pixels w0=16*wid..+15.
__global__ void conv_att_kernel(const float* __restrict__ pmax,
                                const float* __restrict__ pavg,
                                const float* __restrict__ w1,
                                const float* __restrict__ w2,
                                const float* __restrict__ w3,
                                const float* __restrict__ g1, const float* __restrict__ be1,
                                const float* __restrict__ m1, const float* __restrict__ v1,
                                const float* __restrict__ g2, const float* __restrict__ be2,
                                const float* __restrict__ m2, const float* __restrict__ v2,
                                const float* __restrict__ g3, const float* __restrict__ be3,
                                const float* __restrict__ m3, const float* __restrict__ v3,
                                float* __restrict__ att) {
  // lp[ch][r][col]: rows h-3..h+3 in r=0..6, r=7 stays zero; col = w+3, cols
  // 0..2 and 67..71 stay zero (padding). Flat stride 72, channel stride 576.
  __shared__ float lp[2][8][72];
  __shared__ float btabT[16][128];  // B transposed: [n][k], k = ch*64+r*8+s
  __shared__ float ytmp[4][3][16];  // per-wave conv results, 3 branches x 16 px
  const int bh = blockIdx.x;
  const int b = bh >> 6, h = bh & 63;
  const int tid = threadIdx.x;      // blockDim = 128

  // Build transposed weight table (zero at n>=3, r==7, s==7).
  for (int idx = tid; idx < 16 * 128; idx += 128) {
    int n = idx >> 7, k = idx & 127;
    int ch = k >> 6, r = (k >> 3) & 7, s = k & 7;
    float val = 0.f;
    if (n < 3 && r < 7 && s < 7) {
      if (n == 0)      val = w1[ch * 49 + r * 7 + s];
      else if (n == 1) val = w2[ch * 49 + s * 7 + r];  // transposed taps
      else             val = w3[ch * 49 + r * 7 + s];
    }
    btabT[n][k] = val;
  }
  // Zero-pad pooled tile, then fill valid region.
  for (int idx = tid; idx < 2 * 8 * 72; idx += 128) ((float*)lp)[idx] = 0.f;
  __syncthreads();
  for (int idx = tid; idx < 2 * 7 * 64; idx += 128) {
    int w = idx & 63;
    int rr = (idx >> 6) % 7;
    int ch = idx / (7 * 64);
    int hh = h + rr - 3;
    if (hh >= 0 && hh < 64) {
      const float* src = (ch == 0) ? pmax : pavg;
      lp[ch][rr][w + 3] = src[(b * 64 + hh) * 64 + w];
    }
  }
  __syncthreads();

  const int lane = tid & 31, wid = tid >> 5;
  const int m = lane & 15, khalf = lane >> 4;  // A: lane = M + 16*(K/2)
  const int w0 = wid * 16;
  const float* lpf = &lp[0][0][0];
  const float* btf = &btabT[0][0];
  const int aBase = w0 + m + 2 * khalf;        // lane part of A LDS index
  const int bBase = m * 128 + 2 * khalf;       // lane part of B LDS index (even)

  v8f acc = {};
#if __has_builtin(__builtin_amdgcn_wmma_f32_16x16x4_f32)
#pragma unroll
  for (int chunk = 0; chunk < 32; ++chunk) {
    const int k0 = chunk * 4;                         // compile-time
    const int ch = k0 >> 6;                           // compile-time
    const int r  = (k0 >> 3) & 7;                     // compile-time (no carry)
    const int sc = k0 & 7;                            // compile-time (0 or 4)
    const int ai = ch * 576 + r * 72 + sc + aBase;    // base + immediates
    v2f a;
    a[0] = lpf[ai];
    a[1] = lpf[ai + 1];
    v2f bm = *(const v2f*)(btf + bBase + k0);         // 8B-aligned pair
    // D(16x16,f32) += A(16x4) x B(4x16)
    acc = __builtin_amdgcn_wmma_f32_16x16x4_f32(
        /*neg_a=*/false, a, /*neg_b=*/false, bm,
        /*c_mod=*/(short)0, acc, /*reuse_a=*/false, /*reuse_b=*/false);
  }
#else
  // Scalar fallback producing the identical C/D register layout.
  for (int k = 0; k < 128; ++k) {
    int ch = k >> 6, r = (k >> 3) & 7, s = k & 7;
    float bw = btf[m * 128 + k];
#pragma unroll
    for (int v = 0; v < 8; ++v) {
      int M = v + 8 * khalf;
      acc[v] += lp[ch][r][w0 + M + s] * bw;
    }
  }
#endif

  // Lane with N=m<3 holds D[M=v+8*khalf][N=m] in acc[v]; scatter via LDS.
  if (m < 3) {
#pragma unroll
    for (int v = 0; v < 8; ++v) ytmp[wid][m][v + 8 * khalf] = acc[v];
  }
  __syncthreads();

  if (lane < 16) {
    float y0 = ytmp[wid][0][lane];
    float y1 = ytmp[wid][1][lane];
    float y2 = ytmp[wid][2][lane];
    float s1 = g1[0] / sqrtf(v1[0] + EPSV), t1 = be1[0] - m1[0] * s1;
    float s2 = g2[0] / sqrtf(v2[0] + EPSV), t2 = be2[0] - m2[0] * s2;
    float s3 = g3[0] / sqrtf(v3[0] + EPSV), t3 = be3[0] - m3[0] * s3;
    float a1 = 1.f / (1.f + expf(-(y0 * s1 + t1)));
    float a2 = 1.f / (1.f + expf(-(y1 * s2 + t2)));
    float a3 = 1.f / (1.f + expf(-(y2 * s3 + t3)));
    att[bh * 64 + w0 + lane] = (a1 + a2 + a3) * (1.f / 3.f);
  }
}

// ---------------- Kernel 3: out = x * att (broadcast over C) ----------------
__global__ void scale_kernel(const float* __restrict__ x,
                             const float* __restrict__ att,
                             float* __restrict__ out) {
  const size_t i4 = (size_t)blockIdx.x * blockDim.x + threadIdx.x;
  const size_t base = i4 * 4;
  const int b = (int)(base >> 20);     // CHW = 1<<20
  const int hw = (int)(base & 4095);   // c-stride is 4096, so hw = base % 4096
  float4 xv = ((const float4*)x)[i4];
  float4 av = *(const float4*)(att + (size_t)b * HW + hw);
  float4 o;
  o.x = xv.x * av.x;
  o.y = xv.y * av.y;
  o.z = xv.z * av.z;
  o.w = xv.w * av.w;
  ((float4*)out)[i4] = o;
}

extern "C" void kernel_launch(void* const* d_in, const int* in_sizes, int n_in,
                              void* d_out, int out_size, void* d_ws, size_t ws_size,
                              hipStream_t stream) {
  const float* x  = (const float*)d_in[0];
  const float* w1 = (const float*)d_in[1];
  const float* g1 = (const float*)d_in[2];
  const float* b1 = (const float*)d_in[3];
  const float* m1 = (const float*)d_in[4];
  const float* v1 = (const float*)d_in[5];
  const float* w2 = (const float*)d_in[6];
  const float* g2 = (const float*)d_in[7];
  const float* b2 = (const float*)d_in[8];
  const float* m2 = (const float*)d_in[9];
  const float* v2 = (const float*)d_in[10];
  const float* w3 = (const float*)d_in[11];
  const float* g3 = (const float*)d_in[12];
  const float* b3 = (const float*)d_in[13];
  const float* m3 = (const float*)d_in[14];
  const float* v3 = (const float*)d_in[15];

  float* ws   = (float*)d_ws;          // needs 3*32*4096*4 = 1.5 MB
  float* pmax = ws;
  float* pavg = ws + BATCH * HW;
  float* attb = ws + 2 * BATCH * HW;

  pool_kernel<<<BATCH * HH, 256, 0, stream>>>(x, pmax, pavg);
  conv_att_kernel<<<BATCH * HH, 128, 0, stream>>>(pmax, pavg, w1, w2, w3,
                                                  g1, b1, m1, v1,
                                                  g2, b2, m2, v2,
                                                  g3, b3, m3, v3, attb);
  const int total4 = BATCH * CHW / 4;  // 8388608
  scale_kernel<<<total4 / 256, 256, 0, stream>>>(x, attb, (float*)d_out);
}